// GCN_interaction_69131793596496
// MI455X (gfx1250) — compile-verified
//
#include <hip/hip_runtime.h>

typedef __attribute__((ext_vector_type(2))) float v2f;
typedef __attribute__((ext_vector_type(8))) float v8f;

#define N_NODES 200000
#define N_EDGES 1000000
#define D 64
#define TILES (N_EDGES / 16)      // 62500
#define WAVES_PER_WG 8
#define WG_SIZE (WAVES_PER_WG * 32)

// ---------------------------------------------------------------------------
// Zero the three accumulator sections of d_out (harness poisons with 0xAA).
// ---------------------------------------------------------------------------
__global__ void zero_kernel(float* __restrict__ out, long n4) {
  float4* o = (float4*)out;
  long stride = (long)gridDim.x * blockDim.x;
  for (long i = (long)blockIdx.x * blockDim.x + threadIdx.x; i < n4; i += stride) {
    o[i] = make_float4(0.f, 0.f, 0.f, 0.f);
  }
}

// ---------------------------------------------------------------------------
// Fused edge pass: per 16-edge tile (one wave32 each):
//   r    = review_feat @ W1^T   (exact fp32 via V_WMMA_F32_16X16X4_F32)
//   r_fe = review_feat @ W2^T
//   atomic scatter: h1 += r*ci_src ; h2 += (feature2[src]+r_fe)*ci_src ;
//                   h3 += feature3[src]*ci_src   (all keyed by dst)
// LDS holds W1^T / W2^T pair-swizzled: entry [m][p*64+n] = {W[n*64+2p], W[n*64+2p+1]}
// so a B fragment for (k0 even, n) is one ds_load_b64.
// ---------------------------------------------------------------------------
__global__ void __launch_bounds__(WG_SIZE)
edge_kernel(const int* __restrict__ src, const int* __restrict__ dst,
            const float* __restrict__ review, const float* __restrict__ ci,
            const float* __restrict__ W1, const float* __restrict__ W2,
            const float* __restrict__ feat2, const float* __restrict__ feat3,
            float* __restrict__ out_rst,    // section 0: h2 accumulator
            float* __restrict__ out_re,     // section 1: h1 accumulator
            float* __restrict__ out_id) {   // section 2: h3 accumulator
  __shared__ v2f ldsB[2][32 * 64];          // 32 KB total

  // Stage W1^T, W2^T into LDS (coalesced global reads: lane index -> p).
  for (int idx = threadIdx.x; idx < 2 * 2048; idx += WG_SIZE) {
    int m = idx >> 11;                      // which weight matrix
    int e = idx & 2047;
    int n = e >> 5;                         // output column (row of W)
    int p = e & 31;                         // K-pair index
    const float* w = m ? W2 : W1;
    v2f val = *(const v2f*)(w + n * 64 + 2 * p);   // {W[n][2p], W[n][2p+1]}
    ldsB[m][p * 64 + n] = val;
  }
  __syncthreads();

  const int wave = threadIdx.x >> 5;
  const int lane = threadIdx.x & 31;
  const int half = lane >> 4;               // 0: lanes 0-15, 1: lanes 16-31
  const int lm   = lane & 15;

  for (int tile = blockIdx.x * WAVES_PER_WG + wave; tile < TILES;
       tile += gridDim.x * WAVES_PER_WG) {
    const int e0 = tile << 4;

    // A fragments: lane holds row (lane&15); K = 4t + 2*half + {0,1} per t.
    const float* arow = review + (size_t)(e0 + lm) * D + 2 * half;
    v2f a[16];
#pragma unroll
    for (int t = 0; t < 16; ++t) a[t] = *(const v2f*)(arow + 4 * t);

    // Per-edge metadata for the 8 rows this lane-half owns (M = 8*half + v).
    const int rbase = e0 + 8 * half;
    int se[8], de[8];
    float cie[8];
#pragma unroll
    for (int v = 0; v < 8; ++v) {
      se[v]  = src[rbase + v];
      de[v]  = dst[rbase + v];
      cie[v] = ci[se[v]];
    }

#pragma unroll
    for (int nt = 0; nt < 4; ++nt) {
      const int n0 = nt * 16;
      v8f c1 = {0.f, 0.f, 0.f, 0.f, 0.f, 0.f, 0.f, 0.f};
      v8f c2 = {0.f, 0.f, 0.f, 0.f, 0.f, 0.f, 0.f, 0.f};
#pragma unroll
      for (int t = 0; t < 16; ++t) {
        // B fragment: k0 = 4t + 2*half, cols n0+lm -> pair p = 2t + half.
        v2f b1 = ldsB[0][(2 * t + half) * 64 + n0 + lm];
        v2f b2 = ldsB[1][(2 * t + half) * 64 + n0 + lm];
        c1 = __builtin_amdgcn_wmma_f32_16x16x4_f32(false, a[t], false, b1,
                                                   (short)0, c1, false, false);
        c2 = __builtin_amdgcn_wmma_f32_16x16x4_f32(false, a[t], false, b2,
                                                   (short)0, c2, false, false);
      }

      const int col = n0 + lm;
#pragma unroll
      for (int v = 0; v < 8; ++v) {
        const float cs   = cie[v];
        const size_t doff = (size_t)de[v] * D + col;
        const size_t soff = (size_t)se[v] * D + col;
        unsafeAtomicAdd(out_re  + doff, c1[v] * cs);                   // pass 1
        unsafeAtomicAdd(out_rst + doff, (feat2[soff] + c2[v]) * cs);   // pass 2
        unsafeAtomicAdd(out_id  + doff, feat3[soff] * cs);             // pass 3
      }
    }
  }
}

// ---------------------------------------------------------------------------
// rst = h * ci_dst for all three sections, in place, float4-vectorized.
// ---------------------------------------------------------------------------
__global__ void finalize_kernel(float* __restrict__ out, const float* __restrict__ ci) {
  const long n4 = (long)N_NODES * D / 4;    // float4s per section
  float4* o0 = (float4*)out;
  float4* o1 = o0 + n4;
  float4* o2 = o1 + n4;
  long stride = (long)gridDim.x * blockDim.x;
  for (long i = (long)blockIdx.x * blockDim.x + threadIdx.x; i < n4; i += stride) {
    float c = ci[i >> 4];                   // 16 float4s per 64-wide row
    float4 a = o0[i];
    a.x *= c; a.y *= c; a.z *= c; a.w *= c;
    o0[i] = a;
    float4 b = o1[i];
    b.x *= c; b.y *= c; b.z *= c; b.w *= c;
    o1[i] = b;
    float4 d = o2[i];
    d.x *= c; d.y *= c; d.z *= c; d.w *= c;
    o2[i] = d;
  }
}

// ---------------------------------------------------------------------------
extern "C" void kernel_launch(void* const* d_in, const int* in_sizes, int n_in,
                              void* d_out, int out_size, void* d_ws, size_t ws_size,
                              hipStream_t stream) {
  const int*   src    = (const int*)  d_in[0];
  const int*   dst    = (const int*)  d_in[1];
  const float* review = (const float*)d_in[2];
  const float* ci_r   = (const float*)d_in[3];
  // d_in[4] = feature : assigned in reference but never used.
  const float* W1     = (const float*)d_in[5];
  const float* W2     = (const float*)d_in[6];
  const float* feat2  = (const float*)d_in[7];
  const float* feat3  = (const float*)d_in[8];

  float* out = (float*)d_out;
  const long sec = (long)N_NODES * D;       // 12.8M floats per section
  float* out_rst = out;                     // h2 * ci_dst
  float* out_re  = out + sec;               // h1 * ci_dst
  float* out_id  = out + 2 * sec;           // h3 * ci_dst

  // 1) zero accumulators
  long n4_all = 3 * sec / 4;
  zero_kernel<<<2048, 256, 0, stream>>>(out, n4_all);

  // 2) fused WMMA edge pass with atomic scatter
  int wgs = (TILES + WAVES_PER_WG - 1) / WAVES_PER_WG;   // 7813
  edge_kernel<<<wgs, WG_SIZE, 0, stream>>>(src, dst, review, ci_r, W1, W2,
                                           feat2, feat3, out_rst, out_re, out_id);

  // 3) scale by ci_dst
  finalize_kernel<<<2048, 256, 0, stream>>>(out, ci_r);
}